// SwitchFocusedLoss_80994493268322
// MI455X (gfx1250) — compile-verified
//
#include <hip/hip_runtime.h>

#define B_DIM 256
#define S_DIM 16384
#define C_DIM 3
#define TOL 5
#define TILE 256
#define CHUNKS_PER_ROW 16
#define CHUNK (S_DIM / CHUNKS_PER_ROW)

// ---- CDNA5 async global->LDS path (guarded; falls back to plain loads) ----
#if defined(__AMDGCN__) && __has_builtin(__builtin_amdgcn_global_load_async_to_lds_b32)
#define HAVE_ASYNC_LDS 1
#endif
#if defined(__AMDGCN__) && __has_builtin(__builtin_amdgcn_global_load_async_to_lds_b128)
#define HAVE_ASYNC_B128 1
#endif

#if defined(HAVE_ASYNC_LDS) || defined(HAVE_ASYNC_B128)
typedef __attribute__((address_space(1))) int as1_int;
typedef __attribute__((address_space(3))) int as3_int;
typedef int v4i __attribute__((ext_vector_type(4)));
typedef __attribute__((address_space(1))) v4i as1_v4i;
typedef __attribute__((address_space(3))) v4i as3_v4i;
#if __has_builtin(__builtin_amdgcn_s_wait_asynccnt)
#define WAIT_ASYNC(n) __builtin_amdgcn_s_wait_asynccnt(n)
#else
#define WAIT_ASYNC(n) asm volatile("s_wait_asynccnt " #n ::: "memory")
#endif
#else
#define WAIT_ASYNC(n)
#endif

// Stage one TILE*3-dword logits tile (16B-aligned base) into LDS.
__device__ __forceinline__ void stage_tile(const float* gsrc, float* lbuf, int tid) {
#if defined(HAVE_ASYNC_B128)
  if (tid < (TILE * C_DIM) / 4) {  // 192 x 16B transfers
    __builtin_amdgcn_global_load_async_to_lds_b128(
        (as1_v4i*)(gsrc + tid * 4), (as3_v4i*)(lbuf + tid * 4),
        /*offset=*/0, /*cpol=*/0);
  }
#elif defined(HAVE_ASYNC_LDS)
#pragma unroll
  for (int k = 0; k < C_DIM; ++k) {
    const int idx = tid + k * TILE;
    __builtin_amdgcn_global_load_async_to_lds_b32(
        (as1_int*)(gsrc + idx), (as3_int*)(lbuf + idx), 0, 0);
  }
#else
#pragma unroll
  for (int k = 0; k < C_DIM; ++k) {
    const int idx = tid + k * TILE;
    lbuf[idx] = gsrc[idx];
  }
#endif
}

__device__ __forceinline__ int argmax3(float a, float b, float c) {
  // first index of maximum (matches jnp.argmax tie-breaking)
  int p = 0;
  float m = a;
  if (b > m) { p = 1; m = b; }
  if (c > m) { p = 2; }
  return p;
}

extern "C" __global__ void switch_loss_init(float* ws) {
  const int i = threadIdx.x;  // 256 threads
  ws[i]        = 0.f;         // loss_sum
  ws[i + 256]  = 0.f;         // extra_sum
  ws[i + 512]  = 0.f;         // valid_sum
  ws[i + 768]  = 0.f;         // true_acc (>0 means row has a true switch)
}

extern "C" __global__ void __launch_bounds__(TILE)
switch_loss_main(const float* __restrict__ logits,
                 const long long* __restrict__ labels,
                 float* __restrict__ ws) {
  const int row   = blockIdx.x / CHUNKS_PER_ROW;
  const int chunk = blockIdx.x % CHUNKS_PER_ROW;
  const int beg   = chunk * CHUNK;
  const int tid   = threadIdx.x;

  __shared__ __align__(16) float lg[2][TILE * C_DIM];  // double-buffered logits
  __shared__ unsigned char psw[TILE + 2 * TOL];
  __shared__ unsigned char tsw[TILE + 2 * TOL];
  __shared__ float4 r4[TILE];

  const size_t row_off = (size_t)row * S_DIM;

  float acc_loss = 0.f, acc_extra = 0.f, acc_valid = 0.f, acc_true = 0.f;

  // Prologue: stage first tile.
  stage_tile(logits + (row_off + (size_t)beg) * C_DIM, lg[0], tid);

  int cur = 0;
  for (int ts = beg; ts < beg + CHUNK; ts += TILE, cur ^= 1) {
    const bool has_next = (ts + TILE < beg + CHUNK);
    // Issue next tile's copy before waiting on the current one (double buffer).
    if (has_next) {
      stage_tile(logits + (row_off + (size_t)(ts + TILE)) * C_DIM, lg[cur ^ 1], tid);
    }

    // Own label (coalesced 8B load, overlaps async copies).
    const int p = ts + tid;
    const long long y = labels[row_off + p];
    if (has_next) {
      __builtin_prefetch(&labels[row_off + p + TILE], 0, 0);
    }

    // Halo switch flags (±TOL positions) recomputed straight from global (L2 hits).
    bool hps = false, hts = false;
    int hidx = -1;
    if (tid < 2 * TOL) {
      const int hp = (tid < TOL) ? (ts - TOL + tid) : (ts + TILE + (tid - TOL));
      hidx = (tid < TOL) ? tid : (TILE + tid);
      if (hp >= 0 && hp < S_DIM) {
        const float* hl = logits + (row_off + (size_t)hp) * C_DIM;
        hps = argmax3(hl[0], hl[1], hl[2]) >= 1;
        hts = labels[row_off + hp] >= 1;
      }
    }

    // Retire only the current tile's copy; the next tile stays in flight.
    if (has_next) { WAIT_ASYNC(1); } else { WAIT_ASYNC(0); }
    __syncthreads();  // all waves' copies of lg[cur] visible

    const float l0 = lg[cur][tid * 3 + 0];
    const float l1 = lg[cur][tid * 3 + 1];
    const float l2 = lg[cur][tid * 3 + 2];
    const bool ps  = argmax3(l0, l1, l2) >= 1;
    const bool trs = (y >= 1);
    psw[tid + TOL] = (unsigned char)ps;
    tsw[tid + TOL] = (unsigned char)trs;
    if (hidx >= 0) { psw[hidx] = (unsigned char)hps; tsw[hidx] = (unsigned char)hts; }
    __syncthreads();

    bool pn = false, tn = false;
#pragma unroll
    for (int k = 0; k <= 2 * TOL; ++k) {
      pn |= (psw[tid + k] != 0);
      tn |= (tsw[tid + k] != 0);
    }

    // weighted CE
    const bool valid = (y != -100);
    const int yi = valid ? (int)y : 0;
    const float m   = fmaxf(l0, fmaxf(l1, l2));
    const float se  = __expf(l0 - m) + __expf(l1 - m) + __expf(l2 - m);
    const float lse = m + __logf(se);
    const float ly  = (yi == 0) ? l0 : ((yi == 1) ? l1 : l2);
    float loss = (lse - ly) * ((yi == 0) ? 0.1f : 5.0f);

    // segmentation adjustment
    loss *= ps ? ((p < S_DIM - 1) ? 0.3f : 3.0f) : 1.0f;
    // proximity reward
    if (trs && pn) loss *= 0.1f;
    if (!valid) loss = 0.f;

    acc_loss += loss;
    if (ps && !tn) acc_extra += loss;   // doubled later iff row has a true switch
    acc_valid += valid ? 1.f : 0.f;
    acc_true += trs ? 1.f : 0.f;

    __syncthreads();  // protect LDS (flags + buffer being refilled next iter)
  }

  // Fused block reduction -> per-row accumulators.
  r4[tid] = make_float4(acc_loss, acc_extra, acc_valid, acc_true);
  __syncthreads();
  for (int s = TILE / 2; s > 0; s >>= 1) {
    if (tid < s) {
      r4[tid].x += r4[tid + s].x;
      r4[tid].y += r4[tid + s].y;
      r4[tid].z += r4[tid + s].z;
      r4[tid].w += r4[tid + s].w;
    }
    __syncthreads();
  }
  if (tid == 0) {
    atomicAdd(&ws[row],        r4[0].x);
    atomicAdd(&ws[row + 256],  r4[0].y);
    atomicAdd(&ws[row + 512],  r4[0].z);
    atomicAdd(&ws[row + 768],  r4[0].w);
  }
}

extern "C" __global__ void switch_loss_final(const float* ws, float* out) {
  __shared__ float s0[256];
  __shared__ float s1[256];
  const int i = threadIdx.x;  // 256 threads, one per row
  const float has_true = ws[i + 768];
  s0[i] = ws[i] + ((has_true > 0.f) ? ws[i + 256] : 0.f);
  s1[i] = ws[i + 512];
  __syncthreads();
  for (int s = 128; s > 0; s >>= 1) {
    if (i < s) { s0[i] += s0[i + s]; s1[i] += s1[i + s]; }
    __syncthreads();
  }
  if (i == 0) out[0] = s0[0] / s1[0];
}

extern "C" void kernel_launch(void* const* d_in, const int* in_sizes, int n_in,
                              void* d_out, int out_size, void* d_ws, size_t ws_size,
                              hipStream_t stream) {
  (void)in_sizes; (void)n_in; (void)out_size; (void)ws_size;
  const float* logits     = (const float*)d_in[0];
  const long long* labels = (const long long*)d_in[1];
  // d_in[2] = tokens: never read (reference only checks it is not None)
  float* ws  = (float*)d_ws;
  float* out = (float*)d_out;

  switch_loss_init<<<dim3(1), dim3(256), 0, stream>>>(ws);
  switch_loss_main<<<dim3(B_DIM * CHUNKS_PER_ROW), dim3(TILE), 0, stream>>>(logits, labels, ws);
  switch_loss_final<<<dim3(1), dim3(256), 0, stream>>>(ws, out);
}